// HyperGCN_56341380989451
// MI455X (gfx1250) — compile-verified
//
#include <hip/hip_runtime.h>

typedef __attribute__((ext_vector_type(2))) float v2f;
typedef __attribute__((ext_vector_type(8))) float v8f;

#define NNODES   100000
#define NEDGES   1600000
#define MIN_NORM 1e-15f
#define MAXN     (1.0f - 1e-5f)
#define WS       68          // padded LDS row stride for X/MX tiles (floats)
#define WPS      66          // padded LDS row stride for W^T k-pair rows (float2)
#define WAVES    4
#define TBS      (WAVES * 32)

__device__ __forceinline__ float artanh_clip(float v) {
    v = fminf(fmaxf(v, -MAXN), MAXN);
    return 0.5f * __logf((1.0f + v) / (1.0f - v));
}

// ---------------------------------------------------------------- prep ------
// h = proj(lorentz_to_poincare(x)), padded to 64 columns (col 63 = 0)
__global__ void prep_kernel(const float* __restrict__ x, float* __restrict__ h, int N) {
    int n = blockIdx.x * blockDim.x + threadIdx.x;
    if (n >= N) return;
    const float* xr = x + (size_t)n * 64;
    float inv = 1.0f / (xr[0] + 1.0f);
    float nn = 0.f;
    float p[63];
#pragma unroll
    for (int c = 0; c < 63; ++c) { p[c] = xr[c + 1] * inv; nn += p[c] * p[c]; }
    nn = fmaxf(sqrtf(nn), MIN_NORM);
    float s = (nn > MAXN) ? MAXN / nn : 1.0f;
    float* hr = h + (size_t)n * 64;
#pragma unroll
    for (int c = 0; c < 63; ++c) hr[c] = s * p[c];
    hr[63] = 0.f;
}

// ------------------------------------------------------- fused dense layer --
// h_out = proj(mobius_add(proj(mobius_matvec(W,x)), expmap0(b)))
// (+ optional relu-act chain, + optional logmap0 output for aggregation layers)
__global__ __launch_bounds__(TBS)
void hyp_dense_kernel(const float* __restrict__ X,   // [N,64] (zero padded)
                      const float* __restrict__ W,   // [outD, inD] row-major
                      const float* __restrict__ bvec,// [outD]
                      float* __restrict__ Y,         // [N, outStride]
                      int N, int inD, int outD, int outStride,
                      int useAct, int outTangent)
{
    __shared__ float2 WsP[32 * WPS];       // W^T, K-pair interleaved: WsP[kp][n] = (Wt[2kp][n], Wt[2kp+1][n])
    __shared__ float  Sx[WAVES][16 * WS];  // per-wave X tile, then MX tile
    __shared__ float  Ybias[64];           // expmap0(b), zero padded
    __shared__ float  yscal[2];            // [0]=tanh(bn)/bn  [1]=|y|^2

    const int tid  = threadIdx.x;
    const int wave = tid >> 5;
    const int lane = tid & 31;
    const int hi   = lane >> 4;     // half-wave select
    const int lr   = lane & 15;

    // --- W^T into LDS, zero-padded, K-pair interleaved for single-b64 B fragments
    for (int i = tid; i < 32 * 64; i += TBS) {
        int kp = i >> 6, n = i & 63;
        int k0 = 2 * kp, k1 = 2 * kp + 1;
        float v0 = (k0 < inD && n < outD) ? W[n * inD + k0] : 0.f;
        float v1 = (k1 < inD && n < outD) ? W[n * inD + k1] : 0.f;
        WsP[kp * WPS + n] = make_float2(v0, v1);
    }
    if (tid == 0) {
        float bn2 = 0.f;
        for (int c = 0; c < outD; ++c) { float v = bvec[c]; bn2 += v * v; }
        float bn = fmaxf(sqrtf(bn2), MIN_NORM);
        float sY = tanhf(bn) / bn;
        yscal[0] = sY;
        yscal[1] = sY * sY * bn2;
    }
    __syncthreads();
    if (tid < 64) Ybias[tid] = (tid < outD) ? yscal[0] * bvec[tid] : 0.f;

    // --- per-wave 16x64 input tile into LDS (rows clamped; keeps EXEC full)
    const int rowBase = (blockIdx.x * WAVES + wave) * 16;
    for (int i = lane; i < 16 * 64; i += 32) {
        int r = i >> 6, c = i & 63;
        int gr = rowBase + r; if (gr >= N) gr = N - 1;
        Sx[wave][r * WS + c] = X[(size_t)gr * 64 + c];
    }
    __syncthreads();

    // --- per-row input norm^2 (lane lr owns row lr of its wave)
    float xn2 = 0.f;
    if (lane < 16) {
        for (int c = 0; c < 64; ++c) { float v = Sx[wave][lr * WS + c]; xn2 += v * v; }
    }

    // --- 16x64 = (16x64) @ (64x64)^T via V_WMMA_F32_16X16X4_F32
    v8f acc[4];
#pragma unroll
    for (int t = 0; t < 4; ++t) { v8f z = {}; acc[t] = z; }
#pragma unroll
    for (int kk = 0; kk < 16; ++kk) {
        const int kb = kk * 4 + 2 * hi;      // A column pair base
        const int kp = kk * 2 + hi;          // W^T k-pair row
        float2 aa = *(const float2*)&Sx[wave][lr * WS + kb];
        v2f a; a[0] = aa.x; a[1] = aa.y;
#pragma unroll
        for (int t = 0; t < 4; ++t) {
            float2 bb = WsP[kp * WPS + t * 16 + lr];
            v2f b; b[0] = bb.x; b[1] = bb.y;
            acc[t] = __builtin_amdgcn_wmma_f32_16x16x4_f32(
                         false, a, false, b, (short)0, acc[t], false, false);
        }
    }
    __syncthreads();            // all A-fragment reads done before overwrite
#pragma unroll
    for (int t = 0; t < 4; ++t)
#pragma unroll
        for (int j = 0; j < 8; ++j)
            Sx[wave][(j + 8 * hi) * WS + t * 16 + lr] = acc[t][j];
    __syncthreads();

    // --- row-wise hyperbolic epilogue (lane lr owns row lr)
    if (lane < 16) {
        float* m = &Sx[wave][lr * WS];
        float xn = fmaxf(sqrtf(xn2), MIN_NORM);
        float mxn2 = 0.f, dotmy = 0.f;
        for (int c = 0; c < outD; ++c) { float v = m[c]; mxn2 += v * v; dotmy += v * Ybias[c]; }
        float mxn = fmaxf(sqrtf(mxn2), MIN_NORM);
        // mobius_matvec scale + proj
        float s1 = tanhf(mxn / xn * artanh_clip(xn)) / mxn;
        float n1 = s1 * mxn;
        if (n1 > MAXN) { s1 *= MAXN / n1; n1 = MAXN; }
        // mobius_add(h, y)
        float x2 = n1 * n1;
        float y2 = yscal[1];
        float xy = s1 * dotmy;
        float cx = 1.f + 2.f * xy + y2;
        float cy = 1.f - x2;
        float den = fmaxf(1.f + 2.f * xy + x2 * y2, MIN_NORM);
        float n2sq = 0.f;
        for (int c = 0; c < outD; ++c) {
            float v = (cx * s1 * m[c] + cy * Ybias[c]) / den;
            m[c] = v; n2sq += v * v;
        }
        float n2 = fmaxf(sqrtf(n2sq), MIN_NORM);
        float s2 = 1.f, nf = n2;
        if (n2 > MAXN) { s2 = MAXN / n2; nf = MAXN; }
        float nfin;
        if (useAct) {
            // proj(expmap0(relu(logmap0(h)))): relu breaks uniformity only in sign
            float sL = artanh_clip(nf) / nf;
            float pos2 = 0.f;
            for (int c = 0; c < outD; ++c) { float v = m[c]; if (v > 0.f) pos2 += v * v; }
            float tn = fmaxf(sL * s2 * sqrtf(pos2), MIN_NORM);
            float th = tanhf(tn);
            float sE = th / tn;
            float sP = (th > MAXN) ? MAXN / th : 1.f;
            float stot = sP * sE * sL * s2;
            for (int c = 0; c < outD; ++c) m[c] = (m[c] > 0.f) ? stot * m[c] : 0.f;
            nfin = fminf(th, MAXN);
        } else {
            for (int c = 0; c < outD; ++c) m[c] *= s2;
            nfin = nf;
        }
        if (outTangent) {   // logmap0 for tangent-space aggregation
            float sT = artanh_clip(nfin) / fmaxf(nfin, MIN_NORM);
            for (int c = 0; c < outD; ++c) m[c] *= sT;
        }
    }
    __syncthreads();

    // --- cooperative coalesced store
    const int lsh = (outD == 64) ? 6 : 5;
    const int msk = outD - 1;
    for (int i = lane; i < (16 << lsh); i += 32) {
        int r = i >> lsh, c = i & msk;
        int gr = rowBase + r;
        if (gr < N) Y[(size_t)gr * outStride + c] = Sx[wave][r * WS + c];
    }
}

// ------------------------------------------------------------ aggregation ---
__global__ void zero_kernel(float* __restrict__ p, size_t n) {
    size_t i = (size_t)blockIdx.x * blockDim.x + threadIdx.x;
    if (i < n) p[i] = 0.f;
}

__global__ void scatter_kernel(const float* __restrict__ t,
                               const long long* __restrict__ ei,  // [2,E] int64
                               float* __restrict__ agg,
                               float* __restrict__ deg, int E)
{
    int tid = blockIdx.x * blockDim.x + threadIdx.x;
    int e = tid >> 4;              // 16 threads per edge, 4 floats each
    int q = tid & 15;
    if (e >= E) return;
    int src = (int)ei[e];
    int dst = (int)ei[(size_t)E + e];
    const float4 v = *(const float4*)(t + (size_t)src * 64 + q * 4);
    float* ad = agg + (size_t)dst * 64 + q * 4;
    atomicAdd(&ad[0], v.x);
    atomicAdd(&ad[1], v.y);
    atomicAdd(&ad[2], v.z);
    atomicAdd(&ad[3], v.w);
    if (q == 0) atomicAdd(&deg[dst], 1.0f);
}

// h = proj(expmap0(agg/deg))
__global__ void post_agg_kernel(const float* __restrict__ agg,
                                const float* __restrict__ deg,
                                float* __restrict__ h, int N)
{
    int n = blockIdx.x * blockDim.x + threadIdx.x;
    if (n >= N) return;
    float inv = 1.0f / fmaxf(deg[n], 1.0f);
    const float* a = agg + (size_t)n * 64;
    float nn = 0.f;
#pragma unroll 8
    for (int c = 0; c < 64; ++c) { float v = a[c] * inv; nn += v * v; }
    nn = fmaxf(sqrtf(nn), MIN_NORM);
    float th = tanhf(nn);
    float s = th / nn;
    if (th > MAXN) s *= MAXN / th;
    s *= inv;
    float* hr = h + (size_t)n * 64;
#pragma unroll 8
    for (int c = 0; c < 64; ++c) hr[c] = s * a[c];
}

// ------------------------------------------------------------------ driver --
extern "C" void kernel_launch(void* const* d_in, const int* in_sizes, int n_in,
                              void* d_out, int out_size, void* d_ws, size_t ws_size,
                              hipStream_t stream)
{
    const float* x  = (const float*)d_in[0];
    const float* W1 = (const float*)d_in[1];  const float* b1 = (const float*)d_in[2];
    const float* W2 = (const float*)d_in[3];  const float* b2 = (const float*)d_in[4];
    const float* W3 = (const float*)d_in[5];  const float* b3 = (const float*)d_in[6];
    const float* W4 = (const float*)d_in[7];  const float* b4 = (const float*)d_in[8];
    const float* W5 = (const float*)d_in[9];  const float* b5 = (const float*)d_in[10];
    const float* W6 = (const float*)d_in[11]; const float* b6 = (const float*)d_in[12];
    const long long* ei = (const long long*)d_in[13];

    const int N = NNODES, E = NEDGES;
    float* hA  = (float*)d_ws;                 // [N,64]
    float* hB  = hA + (size_t)N * 64;          // [N,64]
    float* agg = hB + (size_t)N * 64;          // [N,64]
    float* deg = agg + (size_t)N * 64;         // [N]   (contiguous with agg)

    const dim3 db(TBS);
    const int denseBlocks = (N + WAVES * 16 - 1) / (WAVES * 16);
    const int nodeBlocks  = (N + 255) / 256;
    const int scatBlocks  = (E * 16 + 255) / 256;
    const size_t aggCnt   = (size_t)N * 64 + N;

    prep_kernel<<<nodeBlocks, 256, 0, stream>>>(x, hA, N);

    hyp_dense_kernel<<<denseBlocks, db, 0, stream>>>(hA, W1, b1, hB, N, 63, 64, 64, 1, 0);
    hyp_dense_kernel<<<denseBlocks, db, 0, stream>>>(hB, W2, b2, hA, N, 64, 64, 64, 1, 0);
    hyp_dense_kernel<<<denseBlocks, db, 0, stream>>>(hA, W3, b3, hB, N, 64, 64, 64, 1, 0);

    // layer 4: dense -> tangent, scatter-mean, expmap back
    hyp_dense_kernel<<<denseBlocks, db, 0, stream>>>(hB, W4, b4, hA, N, 64, 64, 64, 1, 1);
    zero_kernel<<<(int)((aggCnt + 255) / 256), 256, 0, stream>>>(agg, aggCnt);
    scatter_kernel<<<scatBlocks, 256, 0, stream>>>(hA, ei, agg, deg, E);
    post_agg_kernel<<<nodeBlocks, 256, 0, stream>>>(agg, deg, hB, N);

    // layer 5
    hyp_dense_kernel<<<denseBlocks, db, 0, stream>>>(hB, W5, b5, hA, N, 64, 64, 64, 1, 1);
    zero_kernel<<<(int)((aggCnt + 255) / 256), 256, 0, stream>>>(agg, aggCnt);
    scatter_kernel<<<scatBlocks, 256, 0, stream>>>(hA, ei, agg, deg, E);
    post_agg_kernel<<<nodeBlocks, 256, 0, stream>>>(agg, deg, hB, N);

    // layer 6: outD=32, no act, straight into d_out
    hyp_dense_kernel<<<denseBlocks, db, 0, stream>>>(hB, W6, b6, (float*)d_out,
                                                     N, 64, 32, 32, 0, 0);
}